// BlockCausalAttention_65231963292115
// MI455X (gfx1250) — compile-verified
//
#include <hip/hip_runtime.h>
#include <hip/hip_bf16.h>

// Problem constants (B=2, T=8, N_p=256, D=1024, H=16, hd=64, L=2048)
#define BB 2
#define LT 2048
#define DD 1024
#define HH 16
#define HD 64

typedef __attribute__((ext_vector_type(2))) float v2f;
typedef __attribute__((ext_vector_type(8))) float v8f;

__device__ __forceinline__ v8f wmma4(v2f a, v2f b, v8f c) {
  // D = A(16x4,f32) * B(4x16,f32) + C(16x16,f32)
  return __builtin_amdgcn_wmma_f32_16x16x4_f32(false, a, false, b, (short)0, c,
                                               false, false);
}

// ---------------------------------------------------------------------------
// Kernel 1: qkv = x @ Wqkv^T   (4096x1024) x (3072x1024)^T
// One wave computes a 16x64 tile of Y; results scattered into per-head
// q/k/v buffers laid out [B][H][L][hd].
// ---------------------------------------------------------------------------
__global__ void __launch_bounds__(256)
qkv_gemm_kernel(const float* __restrict__ X, const float* __restrict__ W,
                float* __restrict__ Qb, float* __restrict__ Kb,
                float* __restrict__ Vb) {
  const int lane = threadIdx.x & 31;
  const int wave = threadIdx.x >> 5;
  const int colBase = (blockIdx.x * 8 + wave) * 64;  // 0..3008, 64-aligned
  const int rowBase = blockIdx.y * 16;               // 0..4080
  const int half = lane >> 4;
  const int l16 = lane & 15;
  const int kph = half << 1;  // K phase: lanes 0-15 -> k0,k1 ; 16-31 -> k2,k3

  const float* xr = X + (size_t)(rowBase + l16) * DD + kph;
  const float* w0 = W + (size_t)(colBase + l16) * DD + kph;

  v8f acc[4] = {};
#pragma unroll 4
  for (int k = 0; k < DD; k += 4) {
    float2 af = *(const float2*)(xr + k);
    v2f a;
    a.x = af.x;
    a.y = af.y;
#pragma unroll
    for (int d = 0; d < 4; ++d) {
      float2 bf = *(const float2*)(w0 + (size_t)d * 16 * DD + k);
      v2f b;
      b.x = bf.x;
      b.y = bf.y;
      acc[d] = wmma4(a, b, acc[d]);
    }
  }

  // colBase is 64-aligned -> whole tile lands in one (q|k|v, head) slice
  const int c = colBase >> 10;          // 0:q 1:k 2:v
  const int h = (colBase & 1023) >> 6;  // head
  float* dst = (c == 0) ? Qb : (c == 1) ? Kb : Vb;
#pragma unroll
  for (int r = 0; r < 8; ++r) {
    const int i = rowBase + r + half * 8;  // row of Y = b*L + l
    const int b = i >> 11;
    const int l = i & (LT - 1);
    const size_t rbase = (((size_t)b * HH + h) * LT + l) * HD;
#pragma unroll
    for (int d = 0; d < 4; ++d) dst[rbase + d * 16 + l16] = acc[d][r];
  }
}

// ---------------------------------------------------------------------------
// Kernel 2: in-place RMSNorm (over hd=64) + RoPE on q or k.
// One wave per (b,h,l) row; lane j owns the interleaved pair (2j, 2j+1).
// ---------------------------------------------------------------------------
__global__ void __launch_bounds__(256)
norm_rope_kernel(float* __restrict__ buf, const float* __restrict__ scale) {
  const int wave = threadIdx.x >> 5;
  const int lane = threadIdx.x & 31;
  const size_t row = (size_t)blockIdx.x * 8 + wave;  // over B*H*L rows
  const int l = (int)(row & (LT - 1));

  float2* p = (float2*)(buf + row * HD) + lane;
  float2 v = *p;
  float ss = v.x * v.x + v.y * v.y;
#pragma unroll
  for (int off = 16; off; off >>= 1) ss += __shfl_xor(ss, off, 32);
  const float r = rsqrtf(ss * (1.0f / 64.0f) + 1e-6f);
  const float xr = v.x * r * scale[2 * lane];
  const float xi = v.y * r * scale[2 * lane + 1];

  // angles: j<16 -> temporal (l/256), j>=16 -> spatial (l%256)
  const int jj = (lane < 16) ? lane : (lane - 16);
  const float pos = (lane < 16) ? (float)(l >> 8) : (float)(l & 255);
  const float invf = expf((float)jj * (-9.210340371976184f / 16.0f));
  const float ang = pos * invf;
  const float c = cosf(ang), s = sinf(ang);
  float2 o;
  o.x = xr * c - xi * s;
  o.y = xr * s + xi * c;
  *p = o;
}

// ---------------------------------------------------------------------------
// Kernel 3: frame-causal flash attention. One wave per 16-query tile per
// (b,h). Output written as [B][L][H*hd] for the final projection.
// ---------------------------------------------------------------------------
__global__ void __launch_bounds__(32)
attn_kernel(const float* __restrict__ Qb, const float* __restrict__ Kb,
            const float* __restrict__ Vb, float* __restrict__ Ab) {
  const int lane = threadIdx.x;
  const int half = lane >> 4;
  const int l16 = lane & 15;
  const int kph = half << 1;
  const int q0 = blockIdx.x * 16;
  const int bh = blockIdx.y;
  const int nkey = ((q0 >> 8) + 1) << 8;  // keys with frame <= query frame
  const size_t base = (size_t)bh * LT * HD;

  __shared__ float sP[16 * 16];

  // Preload all 16 A-fragments of the Q tile (16 rows x 64 dims)
  v2f qa[16];
  const float* qrow = Qb + base + (size_t)(q0 + l16) * HD + kph;
#pragma unroll
  for (int c = 0; c < 16; ++c) {
    float2 t = *(const float2*)(qrow + 4 * c);
    qa[c].x = t.x;
    qa[c].y = t.y;
  }

  v8f o0 = {}, o1 = {}, o2 = {}, o3 = {};
  float m[8], lsum[8];
#pragma unroll
  for (int r = 0; r < 8; ++r) {
    m[r] = -3.0e38f;
    lsum[r] = 0.0f;
  }

  for (int key0 = 0; key0 < nkey; key0 += 16) {
    // S = Q_tile . K_chunk^T   (16 WMMAs over hd=64)
    const float* krow = Kb + base + (size_t)(key0 + l16) * HD + kph;
    v8f s = {};
#pragma unroll
    for (int c = 0; c < 16; ++c) {
      float2 t = *(const float2*)(krow + 4 * c);
      v2f b;
      b.x = t.x;
      b.y = t.y;
      s = wmma4(qa[c], b, s);
    }

    // Online softmax; C-layout row r lives in vgpr r of each 16-lane half
    float fac[8];
#pragma unroll
    for (int r = 0; r < 8; ++r) {
      const float sv = s[r] * 0.125f;  // hd^-0.5
      float mx = sv;
#pragma unroll
      for (int off = 1; off < 16; off <<= 1)
        mx = fmaxf(mx, __shfl_xor(mx, off, 32));
      const float nm = fmaxf(m[r], mx);
      const float p = __expf(sv - nm);
      s[r] = p;
      float rs = p;
#pragma unroll
      for (int off = 1; off < 16; off <<= 1) rs += __shfl_xor(rs, off, 32);
      fac[r] = __expf(m[r] - nm);
      lsum[r] = lsum[r] * fac[r] + rs;
      m[r] = nm;
    }
#pragma unroll
    for (int r = 0; r < 8; ++r) {
      o0[r] *= fac[r];
      o1[r] *= fac[r];
      o2[r] *= fac[r];
      o3[r] *= fac[r];
    }

    // Reshape P: C-layout -> A-layout via LDS (1KB tile)
    __syncthreads();
#pragma unroll
    for (int r = 0; r < 8; ++r) sP[(r + half * 8) * 16 + l16] = s[r];
    __syncthreads();

    // O += P . V_chunk   (4 K-chunks x 4 column blocks of hd)
#pragma unroll
    for (int c = 0; c < 4; ++c) {
      v2f a;
      a.x = sP[l16 * 16 + 4 * c + kph];
      a.y = sP[l16 * 16 + 4 * c + kph + 1];
      const float* vr0 = Vb + base + (size_t)(key0 + 4 * c + kph) * HD;
      const float* vr1 = vr0 + HD;
      v2f b;
      b.x = vr0[l16];      b.y = vr1[l16];      o0 = wmma4(a, b, o0);
      b.x = vr0[16 + l16]; b.y = vr1[16 + l16]; o1 = wmma4(a, b, o1);
      b.x = vr0[32 + l16]; b.y = vr1[32 + l16]; o2 = wmma4(a, b, o2);
      b.x = vr0[48 + l16]; b.y = vr1[48 + l16]; o3 = wmma4(a, b, o3);
    }
  }

  const int bb = bh >> 4, h = bh & 15;
#pragma unroll
  for (int r = 0; r < 8; ++r) {
    const float inv = 1.0f / lsum[r];
    const int l = q0 + r + half * 8;
    float* dst = Ab + ((size_t)bb * LT + l) * DD + h * HD;
    dst[l16] = o0[r] * inv;
    dst[16 + l16] = o1[r] * inv;
    dst[32 + l16] = o2[r] * inv;
    dst[48 + l16] = o3[r] * inv;
  }
}

// ---------------------------------------------------------------------------
// Kernel 4: out = attn_flat @ Wout^T  (4096x1024) x (1024x1024)^T
// ---------------------------------------------------------------------------
__global__ void __launch_bounds__(256)
out_gemm_kernel(const float* __restrict__ A, const float* __restrict__ W,
                float* __restrict__ Y) {
  const int lane = threadIdx.x & 31;
  const int wave = threadIdx.x >> 5;
  const int colBase = (blockIdx.x * 8 + wave) * 64;  // 0..960
  const int rowBase = blockIdx.y * 16;
  const int half = lane >> 4;
  const int l16 = lane & 15;
  const int kph = half << 1;

  const float* ar = A + (size_t)(rowBase + l16) * DD + kph;
  const float* w0 = W + (size_t)(colBase + l16) * DD + kph;

  v8f acc[4] = {};
#pragma unroll 4
  for (int k = 0; k < DD; k += 4) {
    float2 af = *(const float2*)(ar + k);
    v2f a;
    a.x = af.x;
    a.y = af.y;
#pragma unroll
    for (int d = 0; d < 4; ++d) {
      float2 bf = *(const float2*)(w0 + (size_t)d * 16 * DD + k);
      v2f b;
      b.x = bf.x;
      b.y = bf.y;
      acc[d] = wmma4(a, b, acc[d]);
    }
  }

#pragma unroll
  for (int r = 0; r < 8; ++r) {
    const int i = rowBase + r + half * 8;
#pragma unroll
    for (int d = 0; d < 4; ++d)
      Y[(size_t)i * DD + colBase + d * 16 + l16] = acc[d][r];
  }
}

extern "C" void kernel_launch(void* const* d_in, const int* in_sizes, int n_in,
                              void* d_out, int out_size, void* d_ws,
                              size_t ws_size, hipStream_t stream) {
  const float* x = (const float*)d_in[0];     // (B,L,D)
  const float* Wqkv = (const float*)d_in[1];  // (3D,D)
  const float* Wout = (const float*)d_in[2];  // (D,D)
  const float* qs = (const float*)d_in[3];    // (hd,)
  const float* ks = (const float*)d_in[4];    // (hd,)
  float* out = (float*)d_out;                 // (B,L,D)

  const size_t N = (size_t)BB * HH * LT * HD;  // 4M floats per tensor
  float* Qb = (float*)d_ws;                    // [B][H][L][hd]
  float* Kb = Qb + N;
  float* Vb = Kb + N;
  float* Ab = Vb + N;                          // [B][L][D]

  // 1) QKV projection (WMMA f32)
  qkv_gemm_kernel<<<dim3(6, 256), 256, 0, stream>>>(x, Wqkv, Qb, Kb, Vb);
  // 2) RMSNorm + RoPE on q and k
  norm_rope_kernel<<<dim3(8192), 256, 0, stream>>>(Qb, qs);
  norm_rope_kernel<<<dim3(8192), 256, 0, stream>>>(Kb, ks);
  // 3) Frame-causal flash attention (WMMA f32)
  attn_kernel<<<dim3(LT / 16, BB * HH), 32, 0, stream>>>(Qb, Kb, Vb, Ab);
  // 4) Output projection (WMMA f32)
  out_gemm_kernel<<<dim3(2, 256), 256, 0, stream>>>(Ab, Wout, out);
}